// ProductResidualVectorQuantize_57913339020076
// MI455X (gfx1250) — compile-verified
//
#include <hip/hip_runtime.h>

typedef float v2f __attribute__((ext_vector_type(2)));
typedef float v8f __attribute__((ext_vector_type(8)));

__device__ __forceinline__ v8f wmma4(v2f a, v2f b, v8f c) {
  // D = A(16x4) * B(4x16) + C, all f32, wave32 fragment layout.
  return __builtin_amdgcn_wmma_f32_16x16x4_f32(false, a, false, b, (short)0, c, false, false);
}

#define ZQ_ELEMS 50331648ull  /* 16 * 24576 * 128 */
#define XS 1028               /* padded LDS row stride (floats) */

// ---------------- Kernel 0: normalize codebooks ----------------
__global__ void prvq_normalize(const float* __restrict__ embeds,
                               float* __restrict__ en, float* __restrict__ ee) {
  int row = blockIdx.x * 256 + threadIdx.x;   // over 3*6*1024 rows
  if (row >= 3 * 6 * 1024) return;
  float v[8];
  float nr2 = 0.f;
#pragma unroll
  for (int d = 0; d < 8; ++d) { v[d] = embeds[row * 8 + d]; nr2 += v[d] * v[d]; }
  float denom = fmaxf(sqrtf(nr2), 1e-12f);
  float s = 0.f;
#pragma unroll
  for (int d = 0; d < 8; ++d) {
    float e = v[d] / denom;
    en[row * 8 + d] = e;
    s += e * e;
  }
  ee[row] = s;
}

// ---------------- Kernel 0b: build transposed/padded WMMA operand tables ----------------
// pdT[m][n(16)][k(1024)] : proj_down transposed, N-cols 8..15 zero-padded (b64 frag loads)
// puT[m][n(1024)][k(8)]  : proj_up transposed (contiguous K pairs per lane)
__global__ void prvq_prep(const float* __restrict__ pd, const float* __restrict__ pu,
                          float* __restrict__ pdT, float* __restrict__ puT) {
  int idx = blockIdx.x * 256 + threadIdx.x;
  if (idx < 3 * 16 * 1024) {
    int k = idx & 1023;
    int n = (idx >> 10) & 15;
    int m = idx >> 14;
    pdT[idx] = (n < 8) ? pd[((m << 10) + k) * 8 + n] : 0.f;
  }
  if (idx < 3 * 1024 * 8) {
    int k = idx & 7;
    int n = (idx >> 3) & 1023;
    int m = idx >> 13;
    puT[idx] = pu[((m * 8 + k) << 10) + n];
  }
}

// ---------------- Kernel 1: main PRVQ, one 16-token tile per block ----------------
__global__ __launch_bounds__(128) void prvq_main(
    const float* __restrict__ z_e, const float* __restrict__ pdT,
    const float* __restrict__ puT, const float* __restrict__ en,
    const float* __restrict__ ee, float* __restrict__ out,
    float* __restrict__ partial, int ns) {
  __shared__ float xs[16 * XS];          // x slice / zq staging (16 tokens x 1024, padded)
  __shared__ float Rpart[4 * 16 * 8];    // per-wave down-proj partials
  __shared__ float Rs[16 * 9];           // residual (stride 9 -> conflict-free frag loads)
  __shared__ float Zs[16 * 9];           // normalized residual
  __shared__ float Zq[16 * 9];           // accumulated quantized
  __shared__ float zzs[16];
  __shared__ float candV[4 * 16 * 16];
  __shared__ int   candI[4 * 16 * 16];
  __shared__ int   codes_s[16];
  __shared__ float red[128];

  const int tid   = threadIdx.x;
  const int wave  = tid >> 5;
  const int lane  = tid & 31;
  const int lhalf = lane >> 4;   // 0: lanes 0-15, 1: lanes 16-31
  const int lmod  = lane & 15;
  const int tile  = blockIdx.x;  // 1024 tiles
  const int b     = tile >> 6;   // 64 tiles of 16 tokens per batch
  const int t0    = (tile & 63) << 4;

  float lossAcc = 0.f;
  float* codes_out = out + ZQ_ELEMS;

  for (int m = 0; m < 3; ++m) {
    // ---- Phase 1: gather x slice (J in [1024m, 1024m+1024)) into LDS, coalesced ----
    // J = 768*o + h + 6*c ; source = z_e[b][h*4096 + (4*(t0+tk)+o)][c]
    const int oLo = (1024 * m) / 768;
    const int oHi = (1024 * m + 1023) / 768;
    for (int o = oLo; o <= oHi; ++o) {
      for (int h = 0; h < 6; ++h) {
        int base = 768 * o + h - 1024 * m;
        int cLo = (base >= 0) ? 0 : ((-base) + 5) / 6;
        int hi = 1024 - base;
        int cHi = (hi <= 0) ? 0 : (hi + 5) / 6;
        if (cHi > 128) cHi = 128;
        if (cHi <= cLo) continue;
        int c = cLo + tid;
        if (c < cHi) {
          for (int tk = 0; tk < 16; ++tk) {
            int wcol = 4 * (t0 + tk) + o;
            xs[tk * XS + base + 6 * c] =
                z_e[((size_t)b * 24576u + (size_t)h * 4096u + (size_t)wcol) * 128u + (size_t)c];
          }
        }
      }
    }
    __syncthreads();

    // ---- Phase 2: down-proj  R(16x8) = X(16x1024) @ PD(1024x8), K split over waves ----
    // Branch-free: pdT is pre-transposed and zero-padded; two independent WMMA chains.
    {
      const float* pdTm = pdT + (size_t)(((m << 4) + lmod) << 10);
      v8f acc0 = {};
      v8f acc1 = {};
      const int kb = wave << 8;  // 256 K per wave
      for (int k0 = 0; k0 < 256; k0 += 8) {
        int kk = kb + k0 + (lhalf << 1);
        v2f a0, b0, a1, b1;
        a0[0] = xs[lmod * XS + kk];
        a0[1] = xs[lmod * XS + kk + 1];
        b0[0] = pdTm[kk];
        b0[1] = pdTm[kk + 1];
        acc0 = wmma4(a0, b0, acc0);
        int k2 = kk + 4;
        a1[0] = xs[lmod * XS + k2];
        a1[1] = xs[lmod * XS + k2 + 1];
        b1[0] = pdTm[k2];
        b1[1] = pdTm[k2 + 1];
        acc1 = wmma4(a1, b1, acc1);
      }
#pragma unroll
      for (int r = 0; r < 8; ++r)
        if (lmod < 8)
          Rpart[(wave << 7) + (r + (lhalf << 3)) * 8 + lmod] = acc0[r] + acc1[r];
    }
    __syncthreads();
    {
      int tk = tid >> 3, d = tid & 7;
      Rs[tk * 9 + d] = Rpart[tk * 8 + d] + Rpart[128 + tk * 8 + d] +
                       Rpart[256 + tk * 8 + d] + Rpart[384 + tk * 8 + d];
      Zq[tk * 9 + d] = 0.f;
    }
    __syncthreads();

    // ---- Phase 3: sequential residual VQ streams ----
    for (int i = 0; i < ns; ++i) {
      if (tid < 16) {
        float nr2 = 0.f;
#pragma unroll
        for (int d = 0; d < 8; ++d) { float v = Rs[tid * 9 + d]; nr2 += v * v; }
        float denom = fmaxf(sqrtf(nr2), 1e-12f);
        float zz = 0.f;
#pragma unroll
        for (int d = 0; d < 8; ++d) {
          float zv = Rs[tid * 9 + d] / denom;
          Zs[tid * 9 + d] = zv;
          zz += zv * zv;
        }
        zzs[tid] = zz;
      }
      __syncthreads();

      float zzr[8];
#pragma unroll
      for (int r = 0; r < 8; ++r) zzr[r] = zzs[r + (lhalf << 3)];

      v2f A0, A1;  // Z fragment: K 0..3 and 4..7
      A0[0] = Zs[lmod * 9 + (lhalf << 1)];
      A0[1] = Zs[lmod * 9 + (lhalf << 1) + 1];
      A1[0] = Zs[lmod * 9 + 4 + (lhalf << 1)];
      A1[1] = Zs[lmod * 9 + 4 + (lhalf << 1) + 1];

      const float* enb = en + (((size_t)m * 6 + (size_t)i) << 13);
      const float* eeb = ee + (((size_t)m * 6 + (size_t)i) << 10);

      float bestv[8];
      int besti[8];
#pragma unroll
      for (int r = 0; r < 8; ++r) { bestv[r] = 3.4e38f; besti[r] = 0; }

      for (int nt = 0; nt < 16; ++nt) {  // this wave scans codewords [256*wave, +256)
        int n = (wave << 8) + (nt << 4) + lmod;
        v2f B0, B1;  // B[k][n] = En[n][k]  (contiguous b64, L2-resident)
        B0[0] = enb[n * 8 + (lhalf << 1)];
        B0[1] = enb[n * 8 + (lhalf << 1) + 1];
        B1[0] = enb[n * 8 + 4 + (lhalf << 1)];
        B1[1] = enb[n * 8 + 5 + (lhalf << 1)];
        v8f c = {};
        c = wmma4(A0, B0, c);
        c = wmma4(A1, B1, c);
        float eev = eeb[n];
#pragma unroll
        for (int r = 0; r < 8; ++r) {
          float dv = (zzr[r] - 2.0f * c[r]) + eev;  // same association as reference
          if (dv < bestv[r]) { bestv[r] = dv; besti[r] = n; }
        }
      }
#pragma unroll
      for (int r = 0; r < 8; ++r) {
        int row = r + (lhalf << 3);
        candV[((wave << 4) + row) * 16 + lmod] = bestv[r];
        candI[((wave << 4) + row) * 16 + lmod] = besti[r];
      }
      __syncthreads();

      if (tid < 16) {  // exact argmin with first-index tie-break
        float bv = 3.5e38f;
        int bi = 1 << 30;
        for (int s = 0; s < 64; ++s) {
          int wv = s >> 4, ln = s & 15;
          float v = candV[((wv << 4) + tid) * 16 + ln];
          int ix = candI[((wv << 4) + tid) * 16 + ln];
          if (v < bv || (v == bv && ix < bi)) { bv = v; bi = ix; }
        }
        codes_s[tid] = bi;
        codes_out[(((size_t)b * (size_t)ns + (size_t)i) * 3 + (size_t)m) * 1024 +
                  (size_t)(t0 + tid)] = (float)bi;
      }
      __syncthreads();

      {  // residual / zq / loss update
        int tk = tid >> 3, d = tid & 7;
        float q = enb[codes_s[tk] * 8 + d];
        float ro = Rs[tk * 9 + d];
        float df = q - ro;
        lossAcc += df * df;           // cb and cm losses are numerically identical
        Rs[tk * 9 + d] = ro - q;
        Zq[tk * 9 + d] += q;
      }
      __syncthreads();
    }

    // ---- Phase 4: up-proj  Y(16x1024) = Zq(16x8) @ PU(8x1024), N split over waves ----
    {
      v2f A0, A1;
      A0[0] = Zq[lmod * 9 + (lhalf << 1)];
      A0[1] = Zq[lmod * 9 + (lhalf << 1) + 1];
      A1[0] = Zq[lmod * 9 + 4 + (lhalf << 1)];
      A1[1] = Zq[lmod * 9 + 5 + (lhalf << 1)];
      const float* puTm = puT + ((size_t)m << 13);
      for (int nt = 0; nt < 16; ++nt) {
        int n = (wave << 8) + (nt << 4) + lmod;
        v2f B0, B1;  // B[k][n] = puT[m][n][k]  (contiguous b64)
        B0[0] = puTm[n * 8 + (lhalf << 1)];
        B0[1] = puTm[n * 8 + (lhalf << 1) + 1];
        B1[0] = puTm[n * 8 + 4 + (lhalf << 1)];
        B1[1] = puTm[n * 8 + 5 + (lhalf << 1)];
        v8f c = {};
        c = wmma4(A0, B0, c);
        c = wmma4(A1, B1, c);
#pragma unroll
        for (int r = 0; r < 8; ++r) xs[(r + (lhalf << 3)) * XS + n] = c[r];
      }
    }
    __syncthreads();

    // ---- Phase 5: scatter zq back to z_e layout (coalesced, mirrors the gather) ----
    for (int o = oLo; o <= oHi; ++o) {
      for (int h = 0; h < 6; ++h) {
        int base = 768 * o + h - 1024 * m;
        int cLo = (base >= 0) ? 0 : ((-base) + 5) / 6;
        int hi = 1024 - base;
        int cHi = (hi <= 0) ? 0 : (hi + 5) / 6;
        if (cHi > 128) cHi = 128;
        if (cHi <= cLo) continue;
        int c = cLo + tid;
        if (c < cHi) {
          for (int tk = 0; tk < 16; ++tk) {
            int wcol = 4 * (t0 + tk) + o;
            out[((size_t)b * 24576u + (size_t)h * 4096u + (size_t)wcol) * 128u + (size_t)c] =
                xs[tk * XS + base + 6 * c];
          }
        }
      }
    }
    __syncthreads();  // xs reused by next m
  }

  // ---- deterministic per-block loss partial ----
  red[tid] = lossAcc;
  __syncthreads();
  if (tid == 0) {
    float s = 0.f;
    for (int k = 0; k < 128; ++k) s += red[k];
    partial[blockIdx.x] = s;
  }
}

// ---------------- Kernel 2: deterministic loss finalize ----------------
__global__ void prvq_finalize(const float* __restrict__ partial, float* __restrict__ out,
                              int ns) {
  if (blockIdx.x == 0 && threadIdx.x == 0) {
    float s = 0.f;
    for (int k = 0; k < 1024; ++k) s += partial[k];  // fixed order -> deterministic
    float val = (s / 131072.0f) / 3.0f;              // mean over (B,T,8), / NUM_PVQS
    size_t off = ZQ_ELEMS + (size_t)16 * (size_t)ns * 3u * 1024u;
    out[off] = val;      // cb_loss
    out[off + 1] = val;  // cm_loss
  }
}

extern "C" void kernel_launch(void* const* d_in, const int* in_sizes, int n_in,
                              void* d_out, int out_size, void* d_ws, size_t ws_size,
                              hipStream_t stream) {
  (void)in_sizes; (void)n_in; (void)ws_size;
  const float* z_e    = (const float*)d_in[0];
  const float* embeds = (const float*)d_in[1];
  const float* pd     = (const float*)d_in[2];
  const float* pu     = (const float*)d_in[3];
  float* out = (float*)d_out;

  // Recover num_streams on host from out_size: out = zq + B*ns*3*T codes + 2 losses.
  long long ns_ll = ((long long)out_size - (long long)ZQ_ELEMS - 2LL) / (16LL * 3LL * 1024LL);
  int ns = (int)ns_ll;
  if (ns < 0) ns = 0;
  if (ns > 6) ns = 6;

  float* en_ws   = (float*)d_ws;                 // 3*6*1024*8 = 147456 floats
  float* ee_ws   = en_ws + 3 * 6 * 1024 * 8;     // 18432 floats
  float* partial = ee_ws + 3 * 6 * 1024;         // 1024 floats
  float* pdT     = partial + 1024;               // 3*16*1024 = 49152 floats
  float* puT     = pdT + 3 * 16 * 1024;          // 3*1024*8 = 24576 floats

  prvq_normalize<<<72, 256, 0, stream>>>(embeds, en_ws, ee_ws);
  prvq_prep<<<192, 256, 0, stream>>>(pd, pu, pdT, puT);
  prvq_main<<<1024, 128, 0, stream>>>(z_e, pdT, puT, en_ws, ee_ws, out, partial, ns);
  prvq_finalize<<<1, 1, 0, stream>>>(partial, out, ns);
}